// model_indRNN_6906307412189
// MI455X (gfx1250) — compile-verified
//
#include <hip/hip_runtime.h>
#include <cmath>

typedef __attribute__((ext_vector_type(2))) float v2f;
typedef __attribute__((ext_vector_type(8))) float v8f;

#define B_    24
#define T_    8192
#define IN_   55
#define H_    110
#define HP    112          // H padded to multiple of 16 (7 N-tiles)
#define OUT_  23
#define MTOT  (B_ * T_)    // 196608 rows = 1536 * 128 exactly
#define XS_STRIDE 116      // LDS row stride: 16B-aligned rows, conflict-free b64

// ---------------------------------------------------------------------------
// GEMM: out[M x HP] = in[M x Kin] * W^T (+ bias) via V_WMMA_F32_16X16X4_F32.
// 256 threads = 8 waves; each wave: one 16-row strip x all 7 column tiles.
//
// Fragment layouts (ISA 7.12.2, 32-bit, wave32):
//   A 16x4 : lane L -> row M=L&15 ; VGPR0 = K 2*(L>>4), VGPR1 = K 2*(L>>4)+1
//   B 4x16 : lane L -> col N=L&15 ; VGPR0 = K 2*(L>>4), VGPR1 = K 2*(L>>4)+1
//   C/D    : lane L -> col N=L&15 ; VGPR v -> row v + 8*(L>>4)
//
// W is staged in LDS pre-paired along K so a B fragment is ONE ds_load_b64:
//   Wf[(k>>1)*HP*2 + n*2 + (k&1)] = W^T[k][n]   (zero padded)
// ---------------------------------------------------------------------------
__global__ __launch_bounds__(256) void indrnn_gemm(
    const float* __restrict__ in, int in_stride, int Kin, int Kpad,
    const float* __restrict__ w, int Kw,
    const float* __restrict__ bias,
    float* __restrict__ out)
{
    __shared__ float Xs[128 * XS_STRIDE];   // 58.0 KB
    __shared__ float Wf[HP * HP];           // 49.0 KB (paired layout, Kpad<=112)
    __shared__ float Bs[HP];

    const int tid      = threadIdx.x;
    const int base_row = blockIdx.x * 128;

    // ---- stage input tile ----
    if (in_stride == Kpad && (Kpad & 3) == 0) {
        // contiguous 128 x Kpad block: vectorized copy (pad cols already zero)
        const int kv = Kpad >> 2;
        const float4* src = reinterpret_cast<const float4*>(in + (size_t)base_row * in_stride);
        for (int i = tid; i < 128 * kv; i += 256) {
            int r = i / kv, c = i - r * kv;
            *reinterpret_cast<float4*>(&Xs[r * XS_STRIDE + 4 * c]) = src[(size_t)r * kv + c];
        }
    } else {
        for (int i = tid; i < 128 * Kpad; i += 256) {
            int r = i / Kpad, c = i - r * Kpad;
            Xs[r * XS_STRIDE + c] = (c < Kin) ? in[(base_row + r) * in_stride + c] : 0.f;
        }
    }
    // ---- stage W^T in K-paired fragment layout ----
    for (int i = tid; i < Kpad * HP; i += 256) {
        int k = i / HP, n = i - k * HP;
        float v = (k < Kw && n < H_) ? w[n * Kw + k] : 0.f;
        Wf[((k >> 1) * HP + n) * 2 + (k & 1)] = v;
    }
    if (tid < HP) Bs[tid] = (tid < H_) ? bias[tid] : 0.f;
    __syncthreads();

    const int wave = tid >> 5;
    const int lane = tid & 31;
    const int half = lane >> 4;
    const int l16  = lane & 15;
    const int m0   = wave * 16;

    v8f acc[7];
    #pragma unroll
    for (int nt = 0; nt < 7; ++nt) {
        float bv = Bs[nt * 16 + l16];
        acc[nt] = (v8f){bv, bv, bv, bv, bv, bv, bv, bv};
    }

    const float* Arow = &Xs[(m0 + l16) * XS_STRIDE + 2 * half];
    const float* Wl   = &Wf[(half * HP + l16) * 2];
    for (int k0 = 0; k0 < Kpad; k0 += 4) {
        // A fragment: one ds_load_b64 (K pair, 8B aligned)
        const v2f a = *reinterpret_cast<const v2f*>(Arow + k0);
        // B fragments: one ds_load_b64 each, directly pair-aligned
        const float* Wk = Wl + (k0 >> 1) * (HP * 2);
        #pragma unroll
        for (int nt = 0; nt < 7; ++nt) {
            const v2f bf = *reinterpret_cast<const v2f*>(Wk + nt * 32);
            acc[nt] = __builtin_amdgcn_wmma_f32_16x16x4_f32(
                false, a, false, bf, (short)0, acc[nt], false, false);
        }
    }

    // Store D: VGPR v -> row m0+v+8*half, lane -> column nt*16 + l16
    #pragma unroll
    for (int nt = 0; nt < 7; ++nt) {
        #pragma unroll
        for (int v = 0; v < 8; ++v) {
            int row = base_row + m0 + v + 8 * half;
            out[(size_t)row * HP + nt * 16 + l16] = acc[nt][v];
        }
    }
}

// ---------------------------------------------------------------------------
// Diagonal-recurrent scan, in place: h_t = relu(pre_t + u * h_{t-1}).
// One block per batch, one thread per padded channel. Pad channels have
// u=0 and pre=0, so they stay exactly 0 -> next layer's K-pad stays valid.
// ---------------------------------------------------------------------------
__global__ __launch_bounds__(128) void indrnn_scan(
    float* __restrict__ buf, const float* __restrict__ u)
{
    const int b = blockIdx.x;
    const int h = threadIdx.x;
    if (h >= HP) return;
    const float uu = (h < H_) ? u[h] : 0.f;
    float hs = 0.f;
    float* p = buf + (size_t)b * T_ * HP + h;
    #pragma unroll 4
    for (int t = 0; t < T_; ++t) {
        float pre = p[(size_t)t * HP];
        hs = fmaxf(fmaf(uu, hs, pre), 0.f);
        p[(size_t)t * HP] = hs;
    }
}

// ---------------------------------------------------------------------------
// Head: logits = h[:, T-1, :110] @ w_out^T + b_out ; log_softmax over 23.
// ---------------------------------------------------------------------------
__global__ __launch_bounds__(256) void indrnn_head(
    const float* __restrict__ hbuf, const float* __restrict__ w_out,
    const float* __restrict__ b_out, float* __restrict__ out)
{
    __shared__ float lg[B_ * OUT_];
    const int tid = threadIdx.x;
    for (int i = tid; i < B_ * OUT_; i += 256) {
        int b = i / OUT_, o = i - b * OUT_;
        const float* hv = hbuf + ((size_t)b * T_ + (T_ - 1)) * HP;
        float s = b_out[o];
        for (int k = 0; k < H_; ++k) s = fmaf(hv[k], w_out[o * H_ + k], s);
        lg[i] = s;
    }
    __syncthreads();
    for (int i = tid; i < B_ * OUT_; i += 256) {
        int b = i / OUT_;
        float mx = -1e30f;
        for (int o = 0; o < OUT_; ++o) mx = fmaxf(mx, lg[b * OUT_ + o]);
        float se = 0.f;
        for (int o = 0; o < OUT_; ++o) se += expf(lg[b * OUT_ + o] - mx);
        out[i] = lg[i] - mx - logf(se);
    }
}

extern "C" void kernel_launch(void* const* d_in, const int* in_sizes, int n_in,
                              void* d_out, int out_size, void* d_ws, size_t ws_size,
                              hipStream_t stream) {
    const float* x     = (const float*)d_in[0];
    const float* w0    = (const float*)d_in[1];
    const float* w1    = (const float*)d_in[2];
    const float* w2    = (const float*)d_in[3];
    const float* w3    = (const float*)d_in[4];
    const float* u     = (const float*)d_in[5];
    const float* bb    = (const float*)d_in[6];
    const float* w_out = (const float*)d_in[7];
    const float* b_out = (const float*)d_in[8];
    float* out = (float*)d_out;

    const size_t bufElems = (size_t)MTOT * HP;   // 22,020,096 floats = 88 MB
    float* bufA = (float*)d_ws;
    float* bufB = bufA + bufElems;

    dim3 ggrid(MTOT / 128), gblk(256);
    dim3 sgrid(B_), sblk(128);

    // layer 0: K = 55 (pad 56), input stride 55 (scalar staging path)
    indrnn_gemm<<<ggrid, gblk, 0, stream>>>(x, IN_, IN_, 56, w0, IN_, bb + 0 * H_, bufA);
    indrnn_scan<<<sgrid, sblk, 0, stream>>>(bufA, u + 0 * H_);
    // layers 1..3: K = 110 (pad 112, pad cols already zero), contiguous staging
    indrnn_gemm<<<ggrid, gblk, 0, stream>>>(bufA, HP, HP, HP, w1, H_, bb + 1 * H_, bufB);
    indrnn_scan<<<sgrid, sblk, 0, stream>>>(bufB, u + 1 * H_);
    indrnn_gemm<<<ggrid, gblk, 0, stream>>>(bufB, HP, HP, HP, w2, H_, bb + 2 * H_, bufA);
    indrnn_scan<<<sgrid, sblk, 0, stream>>>(bufA, u + 2 * H_);
    indrnn_gemm<<<ggrid, gblk, 0, stream>>>(bufA, HP, HP, HP, w3, H_, bb + 3 * H_, bufB);
    indrnn_scan<<<sgrid, sblk, 0, stream>>>(bufB, u + 3 * H_);

    indrnn_head<<<1, 256, 0, stream>>>(bufB, w_out, b_out, out);
}